// LQE_78469052498464
// MI455X (gfx1250) — compile-verified
//
#include <hip/hip_runtime.h>

typedef float v2f __attribute__((ext_vector_type(2)));
typedef float v8f __attribute__((ext_vector_type(8)));

#define BB   16
#define LL   1000
#define KPT  17      // body points
#define CH   17      // feature channels
#define HH   256
#define WW   256
#define INDIM 51     // KPT*(TOPK+1)
#define INPAD 64     // padded K for WMMA
#define HID  64

// ---------------- Phase 1: bilinear grid-sample + top2 + stat -> X[16000][64] ----------------
__global__ __launch_bounds__(256) void lqe_sample_kernel(
    const float* __restrict__ pred_poses,   // [B, L, 2K]
    const float* __restrict__ feat,         // [B, C, H, W]
    float* __restrict__ xbuf)               // [B*L, INPAD], cols 51..63 zeroed
{
    int tid = blockIdx.x * blockDim.x + threadIdx.x;
    const int total = BB * LL * KPT;
    if (tid >= total) return;

    const int k   = tid % KPT;
    const int row = tid / KPT;              // b*L + l
    const int b   = row / LL;

    const float px = pred_poses[row * (2 * KPT) + 2 * k + 0];
    const float py = pred_poses[row * (2 * KPT) + 2 * k + 1];

    // grid = 2*pose-1 ; x = (grid+1)*0.5*W - 0.5 == pose*W - 0.5
    const float x = px * (float)WW - 0.5f;
    const float y = py * (float)HH - 0.5f;
    const float x0f = floorf(x), y0f = floorf(y);
    const int   x0 = (int)x0f,  y0 = (int)y0f;
    const float fx = x - x0f,   fy = y - y0f;

    float w00 = (1.f - fx) * (1.f - fy);
    float w10 = fx * (1.f - fy);
    float w01 = (1.f - fx) * fy;
    float w11 = fx * fy;

    const bool vx0 = (x0 >= 0) && (x0 < WW);
    const bool vx1 = (x0 + 1 >= 0) && (x0 + 1 < WW);
    const bool vy0 = (y0 >= 0) && (y0 < HH);
    const bool vy1 = (y0 + 1 >= 0) && (y0 + 1 < HH);
    if (!(vx0 && vy0)) w00 = 0.f;
    if (!(vx1 && vy0)) w10 = 0.f;
    if (!(vx0 && vy1)) w01 = 0.f;
    if (!(vx1 && vy1)) w11 = 0.f;

    const int xc0 = min(max(x0,     0), WW - 1);
    const int xc1 = min(max(x0 + 1, 0), WW - 1);
    const int yc0 = min(max(y0,     0), HH - 1);
    const int yc1 = min(max(y0 + 1, 0), HH - 1);
    const int o00 = yc0 * WW + xc0;
    const int o10 = yc0 * WW + xc1;
    const int o01 = yc1 * WW + xc0;
    const int o11 = yc1 * WW + xc1;

    const float* __restrict__ fb = feat + (size_t)b * CH * HH * WW;

    float t1 = -3.4e38f, t2 = -3.4e38f;
#pragma unroll
    for (int c = 0; c < CH; ++c) {
        const float* __restrict__ p = fb + (size_t)c * (HH * WW);
        const float v = w00 * p[o00] + w10 * p[o10] + w01 * p[o01] + w11 * p[o11];
        if (v > t1)      { t2 = t1; t1 = v; }
        else if (v > t2) { t2 = v; }
    }

    float* __restrict__ xr = xbuf + (size_t)row * INPAD;
    xr[3 * k + 0] = t1;
    xr[3 * k + 1] = t2;
    xr[3 * k + 2] = 0.5f * (t1 + t2);
    if (k == KPT - 1) {
#pragma unroll
        for (int j = INDIM; j < INPAD; ++j) xr[j] = 0.f;
    }
}

// ---------------- Phase 2: WMMA f32 GEMM [16 x 64] x [64 x 64] + relu + head ----------------
// Wave handles 16 rows. A layout (f32 16x4): lanes 0-15 = M rows, {VGPR0,VGPR1} = {K0,K1},
// upper half-wave = {K2,K3}. B layout mirrors: VGPR0 = rows K0|K2 striped over lanes (N),
// VGPR1 = rows K1|K3. D: lane = N column, VGPR r = row (r + 8*half).
__global__ __launch_bounds__(256) void lqe_mlp_kernel(
    const float* __restrict__ xbuf,    // [16000, 64]
    const float* __restrict__ scores,  // [16000]
    const float* __restrict__ w1,      // [64, 51]
    const float* __restrict__ b1,      // [64]
    const float* __restrict__ w2,      // [64]
    const float* __restrict__ b2,      // [1]
    float* __restrict__ out)           // [16000]
{
    constexpr int LDW = 68;            // padded LDS row stride (floats)
    __shared__ float w1p[HID * LDW];

    for (int idx = threadIdx.x; idx < HID * LDW; idx += 256) {
        const int r = idx / LDW;
        const int c = idx % LDW;
        w1p[idx] = (c < INDIM) ? w1[r * INDIM + c] : 0.f;
    }
    __syncthreads();

    const int lane = threadIdx.x & 31;
    const int wave = threadIdx.x >> 5;
    const int l16  = lane & 15;
    const int half = lane >> 4;
    const int koff = half * 2;
    const int rowBase = (blockIdx.x * 8 + wave) * 16;
    const int arow = rowBase + l16;

    // Load the wave's 16x64 A-tile: 16 K-steps of float2 per lane.
    v2f a[16];
    const float* __restrict__ xp = xbuf + (size_t)arow * INPAD + koff;
#pragma unroll
    for (int kk = 0; kk < 16; ++kk)
        a[kk] = *(const v2f*)(xp + 4 * kk);

    float partial[8];
#pragma unroll
    for (int r = 0; r < 8; ++r) partial[r] = 0.f;

#pragma unroll
    for (int nt = 0; nt < 4; ++nt) {
        const int n = nt * 16 + l16;            // hidden unit handled by this lane
        const float* __restrict__ wrow = &w1p[n * LDW + koff];
        v8f acc = {};
#pragma unroll
        for (int kk = 0; kk < 16; ++kk) {
            const v2f bvec = *(const v2f*)(wrow + 4 * kk);
            acc = __builtin_amdgcn_wmma_f32_16x16x4_f32(
                false, a[kk], false, bvec, (short)0, acc, false, false);
        }
        const float bias = b1[n];
        const float wv   = w2[n];
#pragma unroll
        for (int r = 0; r < 8; ++r) {
            float h = acc[r] + bias;
            h = fmaxf(h, 0.f);                  // relu
            partial[r] = fmaf(h, wv, partial[r]);
        }
    }

    const float b2v = b2[0];
#pragma unroll
    for (int r = 0; r < 8; ++r) {
        float p = partial[r];
        p += __shfl_xor(p, 8, 16);
        p += __shfl_xor(p, 4, 16);
        p += __shfl_xor(p, 2, 16);
        p += __shfl_xor(p, 1, 16);
        if (l16 == 0) {
            const int m = rowBase + half * 8 + r;
            out[m] = scores[m] + p + b2v;
        }
    }
}

extern "C" void kernel_launch(void* const* d_in, const int* in_sizes, int n_in,
                              void* d_out, int out_size, void* d_ws, size_t ws_size,
                              hipStream_t stream) {
    const float* scores = (const float*)d_in[0];   // [16,1000,1]
    const float* poses  = (const float*)d_in[1];   // [16,1000,34]
    const float* feat   = (const float*)d_in[2];   // [16,17,256,256]
    const float* w1     = (const float*)d_in[3];   // [64,51]
    const float* b1     = (const float*)d_in[4];   // [64]
    const float* w2     = (const float*)d_in[5];   // [1,64]
    const float* b2     = (const float*)d_in[6];   // [1]
    float* out  = (float*)d_out;                   // [16000]
    float* xbuf = (float*)d_ws;                    // needs 16000*64*4 = 4 MB

    const int total = BB * LL * KPT;               // 272000
    lqe_sample_kernel<<<(total + 255) / 256, 256, 0, stream>>>(poses, feat, xbuf);
    lqe_mlp_kernel<<<(BB * LL) / 128, 256, 0, stream>>>(xbuf, scores, w1, b1, w2, b2, out);
}